// GCN_87737591923115
// MI455X (gfx1250) — compile-verified
//
#include <hip/hip_runtime.h>
#include <hip/hip_bf16.h>

// ---------------------------------------------------------------------------
// 2-layer GCN on gfx1250 (MI455X).
//   layer(x, W, b): deg-normalized scatter-add of (x@W) over edges + self-loop,
//                   bias, relu.
// Strategy:
//   * degree/dinv computed once (edge_index identical for both layers)
//   * x@W via V_WMMA_F32_16X16X4_F32 (exact fp32 matrix pipe)
//   * wave-per-edge gather/scale/atomic-scatter (L2 resident, 2 f32 atomics/lane)
//   * self-loop + bias + relu fused into a finalize pass (no atomics for loops)
// ---------------------------------------------------------------------------

typedef __attribute__((ext_vector_type(2))) float v2f;
typedef __attribute__((ext_vector_type(8))) float v8f;

#define DHID 64

__global__ void zero_f32_kernel(float* __restrict__ p, int n) {
    int i = blockIdx.x * blockDim.x + threadIdx.x;
    if (i < n) p[i] = 0.0f;
}

__global__ void degree_kernel(const long long* __restrict__ col,
                              unsigned* __restrict__ deg, int E) {
    int i = blockIdx.x * blockDim.x + threadIdx.x;
    if (i < E) {
        int c = (int)col[i];
        atomicAdd(&deg[c], 1u);
    }
}

// In place: read u32 degree, write float rsqrt(deg + 1)  (+1 = self loop)
__global__ void dinv_kernel(float* __restrict__ degdinv, int n) {
    int i = blockIdx.x * blockDim.x + threadIdx.x;
    if (i < n) {
        unsigned d = ((const unsigned*)degdinv)[i];
        degdinv[i] = rsqrtf((float)(d + 1u));
    }
}

// H[n,64] = X[n,64] @ W[64,64] using V_WMMA_F32_16X16X4_F32.
// 128 threads = 4 waves; wave w owns output cols [16w, 16w+16); block owns
// 16 rows. n must be a multiple of 16 (100000 = 6250*16).
__global__ void gemm64_wmma_kernel(const float* __restrict__ X,
                                   const float* __restrict__ W,
                                   float* __restrict__ H, int n) {
    const int wave = threadIdx.x >> 5;
    const int lane = threadIdx.x & 31;
    const int half = lane >> 4;        // 0: lanes 0-15, 1: lanes 16-31
    const int l16  = lane & 15;
    const int m0   = blockIdx.x * 16;  // row tile
    const int n0   = wave * 16;        // col tile
    if (m0 >= n) return;

    v8f c = {};
    const float* xrow = X + (size_t)(m0 + l16) * DHID;   // A: M = l16
    #pragma unroll
    for (int k0 = 0; k0 < DHID; k0 += 4) {
        // A 16x4 fp32: VGPR0/1 hold K = 2*half, 2*half+1 for row M=l16
        v2f a = *(const v2f*)(xrow + k0 + 2 * half);
        // B 4x16 fp32: VGPR0/1 hold K = 2*half, 2*half+1 for col N=l16
        v2f b;
        b.x = W[(size_t)(k0 + 2 * half + 0) * DHID + n0 + l16];
        b.y = W[(size_t)(k0 + 2 * half + 1) * DHID + n0 + l16];
        c = __builtin_amdgcn_wmma_f32_16x16x4_f32(
                /*neg_a=*/false, a, /*neg_b=*/false, b,
                /*c_mod=*/(short)0, c, /*reuse_a=*/false, /*reuse_b=*/false);
    }
    // C/D: VGPR j -> row m0 + j + 8*half, col n0 + l16
    #pragma unroll
    for (int j = 0; j < 8; ++j) {
        H[(size_t)(m0 + j + 8 * half) * DHID + n0 + l16] = c[j];
    }
}

// One wave per edge: agg[col] += h[row] * dinv[row]*dinv[col]
__global__ void scatter_kernel(const long long* __restrict__ rows,
                               const long long* __restrict__ cols,
                               const float* __restrict__ dinv,
                               const float* __restrict__ h,
                               float* __restrict__ agg, int E) {
    int e    = (blockIdx.x * blockDim.x + threadIdx.x) >> 5;
    int lane = threadIdx.x & 31;
    if (e >= E) return;
    // stream the edge list ahead of the waves (global_prefetch_b8)
    if (e + 4096 < E) {
        __builtin_prefetch(rows + e + 4096, 0, 0);
        __builtin_prefetch(cols + e + 4096, 0, 0);
    }
    int r = (int)rows[e];
    int c = (int)cols[e];
    float nrm = dinv[r] * dinv[c];
    v2f v = *(const v2f*)(h + (size_t)r * DHID + lane * 2);
    float* dst = agg + (size_t)c * DHID + lane * 2;
    atomicAdd(dst + 0, v.x * nrm);
    atomicAdd(dst + 1, v.y * nrm);
}

// In place on agg: agg = relu(agg + h*dinv^2 + bias)   (fuses self-loop msg)
__global__ void finalize_kernel(float* __restrict__ agg,
                                const float* __restrict__ h,
                                const float* __restrict__ dinv,
                                const float* __restrict__ bias, int n) {
    int idx = blockIdx.x * blockDim.x + threadIdx.x;
    if (idx >= n * DHID) return;
    int node = idx >> 6;
    int j    = idx & (DHID - 1);
    float di = dinv[node];
    float v  = agg[idx] + h[idx] * di * di + bias[j];
    agg[idx] = v > 0.0f ? v : 0.0f;
}

extern "C" void kernel_launch(void* const* d_in, const int* in_sizes, int n_in,
                              void* d_out, int out_size, void* d_ws, size_t ws_size,
                              hipStream_t stream) {
    const float*     x    = (const float*)d_in[0];
    const long long* ei   = (const long long*)d_in[1];   // int64 [2, E]
    const float*     W1   = (const float*)d_in[2];
    const float*     b1   = (const float*)d_in[3];
    const float*     W2   = (const float*)d_in[4];
    const float*     b2   = (const float*)d_in[5];
    float*           out  = (float*)d_out;

    const int n = in_sizes[0] / DHID;      // 100000 nodes
    const int E = in_sizes[1] / 2;         // 3,200,000 edges
    const long long* rows = ei;            // edge_index[0] = sources
    const long long* cols = ei + E;        // edge_index[1] = targets

    char*  ws   = (char*)d_ws;
    float* dinv = (float*)(ws);                                  // n (u32 then f32)
    float* h    = (float*)(ws + (size_t)(1u << 20));             // n*64
    float* agg  = (float*)(ws + (size_t)(29u * (1u << 20)));     // n*64

    const int feat  = n * DHID;
    const int TPB   = 256;
    dim3 gN((n + TPB - 1) / TPB), gE((E + TPB - 1) / TPB);
    dim3 gF((feat + TPB - 1) / TPB);
    dim3 gScatter(((size_t)E * 32 + TPB - 1) / TPB);
    dim3 gGemm(n / 16);

    // ---- shared normalization ----
    zero_f32_kernel<<<gN, TPB, 0, stream>>>(dinv, n);
    degree_kernel<<<gE, TPB, 0, stream>>>(cols, (unsigned*)dinv, E);
    dinv_kernel<<<gN, TPB, 0, stream>>>(dinv, n);

    // ---- layer 1 ----
    gemm64_wmma_kernel<<<gGemm, 128, 0, stream>>>(x, W1, h, n);
    zero_f32_kernel<<<gF, TPB, 0, stream>>>(agg, feat);
    scatter_kernel<<<gScatter, TPB, 0, stream>>>(rows, cols, dinv, h, agg, E);
    finalize_kernel<<<gF, TPB, 0, stream>>>(agg, h, dinv, b1, n);   // act1 in agg

    // ---- layer 2 (aggregate straight into d_out) ----
    gemm64_wmma_kernel<<<gGemm, 128, 0, stream>>>(agg, W2, h, n);   // h2 in h
    zero_f32_kernel<<<gF, TPB, 0, stream>>>(out, feat);
    scatter_kernel<<<gScatter, TPB, 0, stream>>>(rows, cols, dinv, h, out, E);
    finalize_kernel<<<gF, TPB, 0, stream>>>(out, h, dinv, b2, n);
}